// MaxUnpooling2D_46840913330177
// MI455X (gfx1250) — compile-verified
//
#include <hip/hip_runtime.h>

// MaxUnpooling2D for B=16, H=128, W=128, C=64, pool 2x2 -> out [16,256,256,64].
// Gather-style formulation: one thread owns one float4 (4 channels) of one
// input element and writes its entire 2x2 output window (value where argmax
// matches the cell's flat per-batch index, zero elsewhere). Every output
// element is written exactly once -> no memset pass, no atomics, traffic is
// the 402 MB roofline minimum. All accesses are B128 and non-temporal
// (gfx1250 th:TH_*_NT) since nothing is re-used.

typedef __attribute__((ext_vector_type(4))) float v4f;
typedef __attribute__((ext_vector_type(4))) int   v4i;

namespace {
constexpr int kB  = 16;
constexpr int kH  = 128;
constexpr int kW  = 128;
constexpr int kC  = 64;
constexpr int kHo = kH * 2;
constexpr int kWo = kW * 2;
constexpr int kC4 = kC / 4;                    // float4 groups per channel dim = 16
constexpr int kN4 = kB * kH * kW * kC4;        // total float4 work items = 4,194,304
}

__global__ __launch_bounds__(256)
void MaxUnpooling2D_46840913330177_kernel(const float* __restrict__ in,
                                          const int*   __restrict__ am,
                                          float*       __restrict__ out) {
    const int tid = blockIdx.x * 256 + threadIdx.x;
    if (tid >= kN4) return;

    // Decode (b, h, w, c4); channel is fastest-moving, all powers of two.
    const int c4 = tid & (kC4 - 1);
    const int w  = (tid >> 4)  & (kW - 1);
    const int h  = (tid >> 11) & (kH - 1);
    const int b  =  tid >> 18;

    // Streaming, read-once inputs -> non-temporal B128 loads.
    const v4f val = __builtin_nontemporal_load((const v4f*)in + tid);
    const v4i idx = __builtin_nontemporal_load((const v4i*)am + tid);

    float* const obase = out + (size_t)b * ((size_t)kHo * kWo * kC);
    const int ch0 = c4 * 4;

    // Write all four cells of this element's 2x2 window. The argmax value is
    // the flat per-batch index (row*Wo + col)*C + ch, so claiming a cell is a
    // direct integer equality test against that cell's flat index.
#pragma unroll
    for (int dh = 0; dh < 2; ++dh) {
#pragma unroll
        for (int dw = 0; dw < 2; ++dw) {
            const int cellbase = ((2 * h + dh) * kWo + (2 * w + dw)) * kC + ch0;
            v4f o;
            o.x = (idx.x == cellbase + 0) ? val.x : 0.0f;
            o.y = (idx.y == cellbase + 1) ? val.y : 0.0f;
            o.z = (idx.z == cellbase + 2) ? val.z : 0.0f;
            o.w = (idx.w == cellbase + 3) ? val.w : 0.0f;
            // Write-once output -> non-temporal B128 store.
            __builtin_nontemporal_store(o, (v4f*)(obase + cellbase));
        }
    }
}

extern "C" void kernel_launch(void* const* d_in, const int* in_sizes, int n_in,
                              void* d_out, int out_size, void* d_ws, size_t ws_size,
                              hipStream_t stream) {
    (void)in_sizes; (void)n_in; (void)out_size; (void)d_ws; (void)ws_size;
    const float* in  = (const float*)d_in[0];
    const int*   am  = (const int*)d_in[1];
    float*       out = (float*)d_out;

    const int blocks = (kN4 + 255) / 256;   // 16384 blocks of 8 wave32s
    MaxUnpooling2D_46840913330177_kernel<<<blocks, 256, 0, stream>>>(in, am, out);
}